// CausalSelfAttention_25950192402665
// MI455X (gfx1250) — compile-verified
//
#include <hip/hip_runtime.h>
#include <hip/hip_bf16.h>
#include <stdint.h>

// ---------------------------------------------------------------------------
// CDNA5 (gfx1250) causal self-attention with top-2 sparsification, fp32.
//   1) qkv = x @ w_attn   (WMMA f32 GEMM, TDM tensor_load_to_lds, double-buffered)
//   2) per-row scores + top-2 + softmax-over-2 + v-gather
//      (VALU kernel, async-to-LDS staging, double-buffered)
//   3) out = y @ w_proj   (same WMMA GEMM)
// Roofline: ~52 GFLOP fp32 vs ~0.2 GB of traffic -> matrix-compute bound;
// all intermediates (48MB qkv) stay resident in the 192MB L2.
// ---------------------------------------------------------------------------

typedef __attribute__((ext_vector_type(2))) float v2f;
typedef __attribute__((ext_vector_type(8))) float v8f;
typedef __attribute__((ext_vector_type(4))) unsigned int u32x4;
typedef __attribute__((ext_vector_type(8))) unsigned int u32x8;

// Per-lane async 16B global->LDS copy (ASYNCcnt path).
#define ASYNC_B128(lds_ptr, g_ptr)                                            \
  asm volatile("global_load_async_to_lds_b128 %0, %1, off"                    \
               :: "v"((unsigned)(uintptr_t)(lds_ptr)),                        \
                  "v"((unsigned long long)(uintptr_t)(g_ptr))                 \
               : "memory")

#define WAIT_ASYNC_0() asm volatile("s_wait_asynccnt 0x0" ::: "memory")
#define WAIT_ASYNC_4() asm volatile("s_wait_asynccnt 0x4" ::: "memory")

// ---------------------------------------------------------------------------
// Tensor Data Mover: 2D tile (tile_h rows x tile_w fp32) from global (row
// pitch stride_elts) into LDS with row padding (D# pad_enable), one wave-level
// instruction. Descriptor per CDNA5 ISA 8.3/8.4:
//   group0: count=1 | lds_addr | global_addr[56:0] | type=2
//   group1: data_size=4B, pad_enable, pad_interval/amount codes,
//           tensor_dim0/1 = tile dims (tile starts at global_addr, no OOB),
//           tile_dim0/1, tensor_dim0_stride = row pitch.
// pad_interval code: interval = 2^(code+1) dwords; pad_amount code: code+1 dwords.
// ---------------------------------------------------------------------------
__device__ __forceinline__ void tdm_load_2d(const float* gaddr, unsigned lds_byte,
                                            int tile_w, int tile_h, int stride_elts,
                                            unsigned pad_interval_code,
                                            unsigned pad_amount_code) {
  unsigned long long ga = (unsigned long long)(uintptr_t)gaddr;
  u32x4 g0;
  g0[0] = 1u;                                             // count=1 (valid user D#)
  g0[1] = lds_byte;                                       // lds_addr
  g0[2] = (unsigned)ga;                                   // global_addr[31:0]
  g0[3] = (unsigned)((ga >> 32) & 0x01FFFFFFull)          // global_addr[56:32]
          | (2u << 30);                                   // type=2 ("image")
  u32x8 g1;
  g1[0] = (2u << 16)                                      // data_size = 4B
        | (1u << 20)                                      // pad_enable
        | (pad_interval_code << 22)
        | (pad_amount_code << 25);                        // workgroup_mask=0
  g1[1] = ((unsigned)tile_w & 0xFFFFu) << 16;             // tensor_dim0[15:0]
  g1[2] = ((unsigned)tile_h & 0xFFFFu) << 16;             // tensor_dim1[15:0]
  g1[3] = ((unsigned)tile_w & 0xFFFFu) << 16;             // tile_dim0
  g1[4] = ((unsigned)tile_h & 0xFFFFu);                   // tile_dim1 (tile_dim2=0)
  g1[5] = (unsigned)stride_elts;                          // tensor_dim0_stride[31:0]
  g1[6] = 0u;
  g1[7] = 0u;
  // 2D tensor: VADDR2/VADDR3 = NULL -> 2-operand form.
  asm volatile("tensor_load_to_lds %0, %1" :: "s"(g0), "s"(g1) : "memory");
}

// ---------------------------------------------------------------------------
// GEMM: C[M,N] = A[M,K] * B[K,N], fp32, V_WMMA_F32_16X16X4_F32.
// 256 threads = 8 waves; block tile 64x64, K-tile 32, double-buffered LDS
// staged by TDM (wave 0 issues, TENSORcnt ping-pong).
// ---------------------------------------------------------------------------
constexpr int BM = 64, BN = 64, BK = 32;
constexpr int LDA_S = BK + 4;  // 36 floats: TDM pad_interval=32dw(code4), pad=4dw(code3)
constexpr int LDB_S = BN + 4;  // 68 floats: TDM pad_interval=64dw(code5), pad=4dw(code3)

__global__ __launch_bounds__(256)
void gemm_wmma_f32(const float* __restrict__ A, const float* __restrict__ B,
                   float* __restrict__ C, int K,
                   int lda, int ldb, int ldc) {
  __shared__ float As[2][BM * LDA_S];
  __shared__ float Bs[2][BK * LDB_S];

  const int tid   = threadIdx.x;
  const int lane  = tid & 31;
  const int wave  = tid >> 5;
  const int tileM = blockIdx.y * BM;
  const int tileN = blockIdx.x * BN;
  const int wm    = (wave & 3) * 16;  // wave row offset inside tile
  const int wn    = (wave >> 2) * 32; // wave col offset inside tile
  const int half  = lane >> 4;        // ISA A/B layout: lanes16-31 hold K+2,K+3
  const int l15   = lane & 15;

  v8f acc0 = {};
  v8f acc1 = {};

  const int nk = K / BK;

  // prologue: prefetch tile 0 into buffer 0 (wave-level TDM, EXEC-independent)
  if (wave == 0) {
    tdm_load_2d(A + (size_t)tileM * lda, (unsigned)(uintptr_t)&As[0][0],
                BK, BM, lda, 4u, 3u);
    tdm_load_2d(B + tileN, (unsigned)(uintptr_t)&Bs[0][0],
                BN, BK, ldb, 5u, 3u);
  }

  for (int i = 0; i < nk; ++i) {
    const int ib = i & 1;
    if (wave == 0) {
      if (i + 1 < nk) {           // prefetch next tile into the other buffer
        const int k0n = (i + 1) * BK;
        tdm_load_2d(A + (size_t)tileM * lda + k0n,
                    (unsigned)(uintptr_t)&As[ib ^ 1][0], BK, BM, lda, 4u, 3u);
        tdm_load_2d(B + (size_t)k0n * ldb + tileN,
                    (unsigned)(uintptr_t)&Bs[ib ^ 1][0], BN, BK, ldb, 5u, 3u);
        __builtin_amdgcn_s_wait_tensorcnt(2);  // drain tile i, keep i+1 in flight
      } else {
        __builtin_amdgcn_s_wait_tensorcnt(0);
      }
    }
    __syncthreads();

    const float* as = &As[ib][0];
    const float* bs = &Bs[ib][0];

#pragma unroll
    for (int kk = 0; kk < BK; kk += 4) {
      const int kb = kk + half * 2;
      v2f a, b0, b1;
      a.x  = as[(wm + l15) * LDA_S + kb];
      a.y  = as[(wm + l15) * LDA_S + kb + 1];
      b0.x = bs[kb * LDB_S + wn + l15];
      b0.y = bs[(kb + 1) * LDB_S + wn + l15];
      b1.x = bs[kb * LDB_S + wn + 16 + l15];
      b1.y = bs[(kb + 1) * LDB_S + wn + 16 + l15];
      acc0 = __builtin_amdgcn_wmma_f32_16x16x4_f32(false, a, false, b0,
                                                   (short)0, acc0, false, false);
      acc1 = __builtin_amdgcn_wmma_f32_16x16x4_f32(false, a, false, b1,
                                                   (short)0, acc1, false, false);
    }
    __syncthreads();   // everyone done reading buf ib before it is refilled
  }

  // epilogue: C/D layout lane=N col, VGPR r = row r + 8*half
#pragma unroll
  for (int r = 0; r < 8; ++r) {
    int row  = tileM + wm + r + half * 8;
    int col0 = tileN + wn + l15;
    C[(size_t)row * ldc + col0]      = acc0[r];
    C[(size_t)row * ldc + col0 + 16] = acc1[r];
  }
}

// ---------------------------------------------------------------------------
// Attention: per-thread query row. Keys staged in double-buffered LDS via
// per-lane async b128 (ASYNCcnt ping-pong: 4 ops/wave/tile, wait <=4).
// Track top-2 (value,index); softmax over the (<=2) causal survivors and
// gather their V rows. Scores scaled 1/sqrt(64)=0.125 (monotone: selection
// unchanged, only exp weights need it). Fully-masked rows -> NaN, matching
// JAX softmax over all -inf.
// ---------------------------------------------------------------------------
constexpr int Tt = 2048, Cc = 1024, Dd = 64, C3 = 3072;

__global__ __launch_bounds__(256)
void attn_top2(const float* __restrict__ qkv, float* __restrict__ y) {
  __shared__ float Ks[2][64 * 68];     // 64 keys x 64 floats, 272B padded rows

  const int tid = threadIdx.x;
  const int b   = blockIdx.z;
  const int h   = blockIdx.y;
  const int t   = blockIdx.x * 256 + tid;

  const float* qrow = qkv + (size_t)(b * Tt + t) * C3 + h * Dd;
  float4 q[16];
#pragma unroll
  for (int i = 0; i < 16; ++i) q[i] = *(const float4*)(qrow + 4 * i);

  float m1 = -INFINITY, m2 = -INFINITY;
  int   i1 = 0, i2 = 0;

  const int ntiles = Tt / 64;

  // stage tile `tile` of keys into buffer `buf` (4 async b128 per wave)
  auto stage = [&](int tile, int buf) {
#pragma unroll
    for (int i = 0; i < 4; ++i) {
      int idx = tid + i * 256;          // 0..1023
      int row = idx >> 4;               // 0..63
      int seg = idx & 15;               // 0..15
      ASYNC_B128(&Ks[buf][row * 68 + seg * 4],
                 qkv + (size_t)(b * Tt + tile * 64 + row) * C3 + Cc + h * Dd + seg * 4);
    }
  };

  stage(0, 0);                          // prologue prefetch

  for (int it = 0; it < ntiles; ++it) {
    const int ib = it & 1;
    if (it + 1 < ntiles) {
      stage(it + 1, ib ^ 1);
      WAIT_ASYNC_4();                   // drain tile it, keep it+1 in flight
    } else {
      WAIT_ASYNC_0();
    }
    __syncthreads();

    const float* ks = &Ks[ib][0];
    for (int jj = 0; jj < 64; ++jj) {   // all lanes same row -> LDS broadcast
      const float* kr = ks + jj * 68;
      float s = 0.f;
#pragma unroll
      for (int i = 0; i < 16; ++i) {
        float4 kv = *(const float4*)(kr + 4 * i);
        s += q[i].x * kv.x + q[i].y * kv.y + q[i].z * kv.z + q[i].w * kv.w;
      }
      const int j = it * 64 + jj;
      if (s > m1)      { m2 = m1; i2 = i1; m1 = s; i1 = j; }
      else if (s > m2) { m2 = s; i2 = j; }
    }
    __syncthreads();
  }

  const float w1  = (i1 <= t) ? 1.0f : 0.0f;
  const float w2  = (i2 <= t) ? __expf((m2 - m1) * 0.125f) : 0.0f;
  const float inv = 1.0f / (w1 + w2);   // no survivors -> inf -> NaN output

  const float* v1 = qkv + (size_t)(b * Tt + i1) * C3 + 2 * Cc + h * Dd;
  const float* v2 = qkv + (size_t)(b * Tt + i2) * C3 + 2 * Cc + h * Dd;
  float* yr = y + (size_t)(b * Tt + t) * Cc + h * Dd;
#pragma unroll
  for (int i = 0; i < 16; ++i) {
    float4 a = *(const float4*)(v1 + 4 * i);
    float4 c = *(const float4*)(v2 + 4 * i);
    float4 o;
    o.x = (w1 * a.x + w2 * c.x) * inv;
    o.y = (w1 * a.y + w2 * c.y) * inv;
    o.z = (w1 * a.z + w2 * c.z) * inv;
    o.w = (w1 * a.w + w2 * c.w) * inv;
    *(float4*)(yr + 4 * i) = o;
  }
}

// ---------------------------------------------------------------------------
// launch: x[2,2048,1024], w_attn[1024,3072], w_proj[1024,1024] -> out[2,2048,1024]
// ws layout: qkv (4096x3072 f32, 48MB) | y (4096x1024 f32, 16MB)
// ---------------------------------------------------------------------------
extern "C" void kernel_launch(void* const* d_in, const int* in_sizes, int n_in,
                              void* d_out, int out_size, void* d_ws, size_t ws_size,
                              hipStream_t stream) {
  const float* x      = (const float*)d_in[0];
  const float* w_attn = (const float*)d_in[1];
  const float* w_proj = (const float*)d_in[2];
  float* out = (float*)d_out;

  float* qkv = (float*)d_ws;                       // 4096 x 3072
  float* y   = qkv + (size_t)4096 * 3072;          // 4096 x 1024

  dim3 blk(256);

  // qkv = x @ w_attn   (M=4096, N=3072, K=1024)
  gemm_wmma_f32<<<dim3(3072 / BN, 4096 / BM), blk, 0, stream>>>(
      x, w_attn, qkv, 1024, 1024, 3072, 3072);

  // attention with top-2 sparsification -> y (head-interleaved [B,T,C])
  attn_top2<<<dim3(Tt / 256, 16, 2), blk, 0, stream>>>(qkv, y);

  // out = y @ w_proj   (M=4096, N=1024, K=1024)
  gemm_wmma_f32<<<dim3(1024 / BN, 4096 / BM), blk, 0, stream>>>(
      y, w_proj, out, 1024, 1024, 1024, 1024);
}